// WaveRNN_68582037782585
// MI455X (gfx1250) — compile-verified
//
#include <hip/hip_runtime.h>

typedef float v2f __attribute__((ext_vector_type(2)));
typedef float v8f __attribute__((ext_vector_type(8)));

#define NZg    256
#define NXg    256
#define NTIME  256
#define NB     8
#define NRECV  64
#define THREADS 1024
#define NWAVES (THREADS / 32)
#define TILES_PER_WAVE (256 / NWAVES)

__device__ __forceinline__ float trid(int i, int j) {
    int d = i - j;
    return (d == 0) ? -2.0f : ((d == 1 || d == -1) ? 1.0f : 0.0f);
}
__device__ __forceinline__ int clampi(int v, int lo, int hi) {
    return v < lo ? lo : (v > hi ? hi : v);
}

// Persistent leapfrog FD kernel: one workgroup per batch, full time loop inside.
// Laplacian interior computed as S*H + H*T via v_wmma_f32_16x16x4_f32 on 16x16 tiles.
__global__ __launch_bounds__(THREADS, 1)
void wave_persistent(const float* __restrict__ x,    // (B, NT)
                     const float* __restrict__ vp,   // (NZ, NX) shared
                     const int*   __restrict__ src,  // (B, 2)
                     const int*   __restrict__ rec,  // (B, NREC, 2)
                     float*       __restrict__ out,  // (B, NT, NREC)
                     float*       __restrict__ ws)   // 2 * B * NZ * NX floats
{
    const int b    = blockIdx.x;
    const int tid  = threadIdx.x;
    const int lane = tid & 31;
    const int wave = tid >> 5;
    const int lm   = lane & 15;   // N (col) index within tile / row index for A
    const int lh   = lane >> 4;   // half-wave selector

    float* fA = ws + (size_t)b        * (NZg * NXg);
    float* fB = ws + (size_t)(NB + b) * (NZg * NXg);

    // Zero-init both ping-pong fields (d_ws is poisoned; must be deterministic).
    for (int i = tid; i < NZg * NXg; i += THREADS) { fA[i] = 0.0f; fB[i] = 0.0f; }
    __builtin_amdgcn_fence(__ATOMIC_RELEASE, "workgroup");
    __syncthreads();
    __builtin_amdgcn_fence(__ATOMIC_ACQUIRE, "workgroup");

    // Constant tridiag(1,-2,1) operator in WMMA A-layout (Sa) and B-layout (Tb).
    // A 16x4 chunk c: lane holds S[lm, 4c+2*lh + {0,1}].
    // B 4x16 chunk c: lane holds T[4c+2*lh + {0,1}, lm].
    v2f Sa[4], Tb[4];
#pragma unroll
    for (int c = 0; c < 4; ++c) {
        const int kb = 4 * c + 2 * lh;
        Sa[c].x = trid(lm, kb);
        Sa[c].y = trid(lm, kb + 1);
        Tb[c].x = trid(kb, lm);
        Tb[c].y = trid(kb + 1, lm);
    }

    const int   sz  = src[b * 2 + 0];
    const int   sx  = src[b * 2 + 1];
    const float dt2 = 1.0e-6f;   // DT*DT
    const float scl = 1.0e-8f;   // DT*DT / (DH*DH)

    for (int t = 0; t < NTIME; ++t) {
        float* h1 = (t & 1) ? fB : fA;   // field at time t
        float* h2 = (t & 1) ? fA : fB;   // field at t-1; hn overwrites it

#pragma unroll 1
        for (int i = 0; i < TILES_PER_WAVE; ++i) {
            const int ti = wave + i * NWAVES;        // 0..255 tile id
            const int z0 = (ti >> 4) << 4;
            const int x0 = (ti & 15) << 4;

            // H tile in A layout: lane holds H[z0+lm, x0+4c+2lh + {0,1}] (8B aligned).
            v2f Ha[4];
#pragma unroll
            for (int c = 0; c < 4; ++c) {
                const float* p = h1 + (size_t)(z0 + lm) * NXg + (x0 + 4 * c + 2 * lh);
                Ha[c] = *(const v2f*)p;
            }
            // H tile in B layout: lane holds H[z0+4c+2lh + {0,1}, x0+lm].
            v2f Hb[4];
#pragma unroll
            for (int c = 0; c < 4; ++c) {
                const int k0 = z0 + 4 * c + 2 * lh;
                Hb[c].x = h1[(size_t)k0       * NXg + (x0 + lm)];
                Hb[c].y = h1[(size_t)(k0 + 1) * NXg + (x0 + lm)];
            }

            // C = S*H + H*T : interior 5-point stencil (sum of 4 nbrs - 4*center).
            v8f C = {0.f, 0.f, 0.f, 0.f, 0.f, 0.f, 0.f, 0.f};
#pragma unroll
            for (int c = 0; c < 4; ++c)
                C = __builtin_amdgcn_wmma_f32_16x16x4_f32(
                        false, Sa[c], false, Hb[c], (short)0, C, false, false);
#pragma unroll
            for (int c = 0; c < 4; ++c)
                C = __builtin_amdgcn_wmma_f32_16x16x4_f32(
                        false, Ha[c], false, Tb[c], (short)0, C, false, false);

            // Halo corrections. C/D layout: VGPR r, lane -> (M = r + 8*lh, N = lm).
            const int gx = x0 + lm;
            int xl = (lm == 0) ? (x0 - 1) : (x0 + 16);   // left halo for N==0, right for N==15
            xl = clampi(xl, 0, NXg - 1);
            const bool lr = (lm == 0) || (lm == 15);
#pragma unroll
            for (int r = 0; r < 8; ++r) {
                const int gz = z0 + r + 8 * lh;
                const float v = h1[(size_t)gz * NXg + xl];
                C[r] += lr ? v : 0.0f;
            }
            {
                int zt = (lh == 0) ? (z0 - 1) : (z0 + 16); // top halo (M==0) / bottom (M==15)
                zt = clampi(zt, 0, NZg - 1);
                const float vt = h1[(size_t)zt * NXg + gx];
                C[0] += (lh == 0) ? vt : 0.0f;
                C[7] += (lh == 1) ? vt : 0.0f;
            }

            // Leapfrog combine; Laplacian forced to zero on the domain border
            // (reference only updates the interior -> border stays exactly 0).
            const bool xe = (gx == 0) || (gx == NXg - 1);
#pragma unroll
            for (int r = 0; r < 8; ++r) {
                const int gz  = z0 + r + 8 * lh;
                const int idx = gz * NXg + gx;
                const float lap = ((gz == 0) || (gz == NZg - 1) || xe) ? 0.0f : C[r];
                const float vv  = vp[idx];
                const float hn  = 2.0f * h1[idx] - h2[idx] + (vv * vv * scl) * lap;
                h2[idx] = hn;   // only this lane touches h2[idx] this step
            }
        }

        __builtin_amdgcn_fence(__ATOMIC_RELEASE, "workgroup");
        __syncthreads();
        __builtin_amdgcn_fence(__ATOMIC_ACQUIRE, "workgroup");

        // Source injection (single thread; hn buffer == h2).
        if (tid == 0) {
            h2[sz * NXg + sx] += dt2 * x[b * NTIME + t];
        }

        __builtin_amdgcn_fence(__ATOMIC_RELEASE, "workgroup");
        __syncthreads();
        __builtin_amdgcn_fence(__ATOMIC_ACQUIRE, "workgroup");

        // Receiver gather: out[b, t, r] = hn[rz, rx].
        if (tid < NRECV) {
            const int rz = rec[(b * NRECV + tid) * 2 + 0];
            const int rx = rec[(b * NRECV + tid) * 2 + 1];
            out[(b * NTIME + t) * NRECV + tid] = h2[rz * NXg + rx];
        }
        // No third barrier needed: next step only writes the old-h1 buffer,
        // which nothing in this region still reads.
    }
}

extern "C" void kernel_launch(void* const* d_in, const int* in_sizes, int n_in,
                              void* d_out, int out_size, void* d_ws, size_t ws_size,
                              hipStream_t stream) {
    const float* x   = (const float*)d_in[0];  // (8, 256) f32
    const float* vp  = (const float*)d_in[1];  // (256, 256) f32
    const int*   src = (const int*)  d_in[2];  // (8, 2) i32
    const int*   rec = (const int*)  d_in[3];  // (8, 64, 2) i32
    float* out = (float*)d_out;                // (8, 256, 64) f32
    float* ws  = (float*)d_ws;                 // needs 2*8*256*256*4 = 4 MB

    wave_persistent<<<NB, THREADS, 0, stream>>>(x, vp, src, rec, out, ws);

    (void)in_sizes; (void)n_in; (void)out_size; (void)ws_size;
}